// SphericalExpansion_25563645346772
// MI455X (gfx1250) — compile-verified
//
#include <hip/hip_runtime.h>
#include <hip/hip_bf16.h>
#include <math.h>

// ---------------- problem constants (from reference) ----------------
constexpr int B       = 8192;
constexpr int D       = 512;
constexpr int NEXP    = 16;
constexpr int TOPK    = 2048;           // round(B * 0.25)
constexpr int NV      = NEXP + 2;       // 18 basis columns
constexpr int NB1     = NEXP + 1;       // 17 (bweights dim)
constexpr int AUGN    = B + TOPK * NEXP; // 40960 output rows

typedef __attribute__((ext_vector_type(2))) float v2f;
typedef __attribute__((ext_vector_type(8))) float v8f;

// Block reduction for 256 threads (8 wave32s), broadcasts result to all threads.
__device__ __forceinline__ float blockReduceSum256(float val, float* red /*>=8*/) {
    #pragma unroll
    for (int off = 16; off > 0; off >>= 1) val += __shfl_down(val, off, 32);
    const int lane = threadIdx.x & 31;
    const int wid  = threadIdx.x >> 5;
    if (lane == 0) red[wid] = val;
    __syncthreads();
    if (wid == 0) {
        float s = (lane < 8) ? red[lane] : 0.0f;
        #pragma unroll
        for (int off = 4; off > 0; off >>= 1) s += __shfl_down(s, off, 32);
        if (lane == 0) red[0] = s;
    }
    __syncthreads();
    float r = red[0];
    __syncthreads();
    return r;
}

// ---------------- kernel 1: normalize x / gathered proxies, scores, raw copy ----
__global__ void se_prep(const float* __restrict__ x, const int* __restrict__ y,
                        const float* __restrict__ prox,
                        float* __restrict__ Xn, float* __restrict__ Wn,
                        float* __restrict__ scores,
                        float* __restrict__ aug_x, float* __restrict__ aug_t) {
    const int b = blockIdx.x;
    const int t = threadIdx.x;
    __shared__ float red[8];

    const size_t ro = (size_t)b * D;
    float x0 = x[ro + t], x1 = x[ro + t + 256];
    aug_x[ro + t] = x0;                 // aug_x[:B] = raw x_batch
    aug_x[ro + t + 256] = x1;

    float ss  = blockReduceSum256(x0 * x0 + x1 * x1, red);
    float xin = 1.0f / sqrtf(ss + 1e-12f);     // torch l2_norm epsilon
    float z0 = x0 * xin, z1 = x1 * xin;
    Xn[ro + t] = z0; Xn[ro + t + 256] = z1;

    const int cls = y[b];
    const size_t po = (size_t)cls * D;
    float p0 = prox[po + t], p1 = prox[po + t + 256];
    float ps  = blockReduceSum256(p0 * p0 + p1 * p1, red);
    float pin = 1.0f / sqrtf(ps + 1e-12f);
    float w0 = p0 * pin, w1 = p1 * pin;
    Wn[ro + t] = w0; Wn[ro + t + 256] = w1;

    float sc = blockReduceSum256(z0 * w0 + z1 * w1, red);
    if (t == 0) { scores[b] = sc; aug_t[b] = (float)cls; }
}

// ---------------- kernel 2: rank-by-count top-k selection flags -----------------
__global__ void se_rank(const float* __restrict__ scores,
                        int* __restrict__ flags, float* __restrict__ is_expan) {
    __shared__ float s[B];                       // 32 KB in LDS
    for (int i = threadIdx.x; i < B; i += blockDim.x) s[i] = scores[i];
    __syncthreads();
    const int b = blockIdx.x * blockDim.x + threadIdx.x;
    const float sb = s[b];
    int cnt = 0;
    for (int j = 0; j < B; ++j) {
        float sj = s[j];
        cnt += (sj > sb) || (sj == sb && j < b);  // jax top_k tie-break
    }
    const int f = (cnt < TOPK) ? 1 : 0;
    flags[b] = f;
    is_expan[b] = f ? 1.0f : 0.0f;
}

// ---------------- kernel 3: prefix-scan flags -> ascending selected list --------
__global__ void se_scan(const int* __restrict__ flags, int* __restrict__ sel) {
    __shared__ int ws[1024];
    const int t = threadIdx.x;
    int f[8]; int sum = 0;
    #pragma unroll
    for (int e = 0; e < 8; ++e) { f[e] = flags[t * 8 + e]; sum += f[e]; }
    ws[t] = sum;
    __syncthreads();
    for (int off = 1; off < 1024; off <<= 1) {   // Hillis–Steele inclusive scan
        int v = (t >= off) ? ws[t - off] : 0;
        __syncthreads();
        ws[t] += v;
        __syncthreads();
    }
    int pos = ws[t] - sum;                       // exclusive prefix
    #pragma unroll
    for (int e = 0; e < 8; ++e) {
        if (f[e]) { if (pos < TOPK) sel[pos] = t * 8 + e; ++pos; }
    }
}

// ---------------- kernel 4: Gram-Schmidt + WMMA f32 expansion GEMM --------------
// One workgroup (256 thr = 8 wave32) per selected sample.
__global__ void se_expand(const float* __restrict__ Xn, const float* __restrict__ Wn,
                          const float* __restrict__ scores, const int* __restrict__ sel,
                          const int* __restrict__ y, const float* __restrict__ rand_units,
                          const float* __restrict__ bweights,
                          float* __restrict__ aug_x, float* __restrict__ aug_t) {
    __shared__ float vS[NV * D];                 // 18x512 basis columns, 36 KB
    __shared__ float Msh[D];                     // projection M = (z.w) w
    __shared__ float bwS[NB1 * NB1];             // 17x17 bweights
    __shared__ float red[8];

    const int k = blockIdx.x;
    const int t = threadIdx.x;
    const int b = sel[k];

    for (int i = t; i < NB1 * NB1; i += 256) bwS[i] = bweights[i];

    const size_t ro = (size_t)b * D;
    float z0 = Xn[ro + t], z1 = Xn[ro + t + 256];
    float w0 = Wn[ro + t], w1 = Wn[ro + t + 256];
    const float s = scores[b];                   // dot(z, w)

    float M0 = s * w0, M1 = s * w1;
    Msh[t] = M0; Msh[t + 256] = M1;
    float r10 = z0 - M0, r11 = z1 - M1;
    float rss = blockReduceSum256(r10 * r10 + r11 * r11, red);
    const float r1n = sqrtf(rss);
    const float rinv = 1.0f / r1n;

    vS[0 * D + t] = w0;          vS[0 * D + t + 256] = w1;         // v0 = w_c
    vS[1 * D + t] = r10 * rinv;  vS[1 * D + t + 256] = r11 * rinv; // v1 = u1
    __syncthreads();

    // Gram-Schmidt: 16 sequential steps; coefficients from ORIGINAL r (matches ref)
    for (int i = 2; i < NV; ++i) {
        const float* rr = rand_units + ((size_t)k * NEXP + (i - 2)) * D;
        const float r0 = rr[t], r1v = rr[t + 256];
        float nb0 = r0, nb1 = r1v;
        for (int j = 0; j < i; ++j) {
            float c = blockReduceSum256(vS[j * D + t] * r0 + vS[j * D + t + 256] * r1v, red);
            nb0 -= c * vS[j * D + t];
            nb1 -= c * vS[j * D + t + 256];
        }
        float nn = blockReduceSum256(nb0 * nb0 + nb1 * nb1, red);
        float ninv = 1.0f / sqrtf(nn);
        vS[i * D + t] = nb0 * ninv;
        vS[i * D + t + 256] = nb1 * ninv;
        __syncthreads();
    }

    // GEMM: u[m][n] = sum_{kk<17} bweights[m+1][kk] * v[1+kk][n], m=0..15, n=0..511
    // V_WMMA_F32_16X16X4_F32: K padded 17->20 (5 chunks of 4). 4 N-tiles per wave.
    const int wave = t >> 5;
    const int lane = t & 31;
    const int l16  = lane & 15;
    const int hi   = lane >> 4;                  // 0: lanes 0-15, 1: lanes 16-31

    for (int tile = wave * 4; tile < wave * 4 + 4; ++tile) {
        const int n0 = tile * 16;
        v8f acc = {};
        #pragma unroll
        for (int ch = 0; ch < 5; ++ch) {
            // A 16x4 f32: VGPR0 = K=(base+2*hi), VGPR1 = K=(base+2*hi+1), M = l16
            const int ka0 = ch * 4 + hi * 2;
            const int ka1 = ka0 + 1;
            v2f A, Bv;
            A.x  = (ka0 < NB1) ? bwS[(l16 + 1) * NB1 + ka0] : 0.0f;
            A.y  = (ka1 < NB1) ? bwS[(l16 + 1) * NB1 + ka1] : 0.0f;
            // B 4x16 f32: rows striped across lanes within a VGPR (mirror of A)
            Bv.x = (ka0 < NB1) ? vS[(1 + ka0) * D + n0 + l16] : 0.0f;
            Bv.y = (ka1 < NB1) ? vS[(1 + ka1) * D + n0 + l16] : 0.0f;
            acc = __builtin_amdgcn_wmma_f32_16x16x4_f32(
                false, A, false, Bv, (short)0, acc, false, false);
        }
        // D layout: VGPR r -> M = r + 8*hi, N = n0 + l16. Epilogue: M + u*r1n.
        #pragma unroll
        for (int r = 0; r < 8; ++r) {
            const int m = r + hi * 8;
            const int n = n0 + l16;
            aug_x[((size_t)B + (size_t)k * NEXP + m) * D + n] = Msh[n] + acc[r] * r1n;
        }
    }
    if (t < NEXP) aug_t[(size_t)B + (size_t)k * NEXP + t] = (float)y[b];
}

// ---------------- launcher ------------------------------------------------------
extern "C" void kernel_launch(void* const* d_in, const int* in_sizes, int n_in,
                              void* d_out, int out_size, void* d_ws, size_t ws_size,
                              hipStream_t stream) {
    const float* x     = (const float*)d_in[0];   // [B, D]
    const int*   yb    = (const int*)  d_in[1];   // [B]
    const float* prox  = (const float*)d_in[2];   // [NB_CLASSES, D]
    const float* bw    = (const float*)d_in[3];   // [17, 17]
    const float* rand_units = (const float*)d_in[4]; // [TOPK, NEXP, D]

    float* aug_x = (float*)d_out;                          // [AUGN, D]
    float* aug_t = aug_x + (size_t)AUGN * D;               // [AUGN]
    float* is_ex = aug_t + AUGN;                           // [B]

    float* Xn     = (float*)d_ws;                          // [B, D]
    float* Wn     = Xn + (size_t)B * D;                    // [B, D]
    float* scores = Wn + (size_t)B * D;                    // [B]
    int*   flags  = (int*)(scores + B);                    // [B]
    int*   selp   = flags + B;                             // [TOPK]

    se_prep  <<<B,        256,  0, stream>>>(x, yb, prox, Xn, Wn, scores, aug_x, aug_t);
    se_rank  <<<B / 256,  256,  0, stream>>>(scores, flags, is_ex);
    se_scan  <<<1,        1024, 0, stream>>>(flags, selp);
    se_expand<<<TOPK,     256,  0, stream>>>(Xn, Wn, scores, selp, yb, rand_units, bw,
                                             aug_x, aug_t);
}